// GINConv_19619410608393
// MI455X (gfx1250) — compile-verified
//
#include <hip/hip_runtime.h>
#include <stddef.h>

typedef __attribute__((ext_vector_type(2))) float v2f;
typedef __attribute__((ext_vector_type(8))) float v8f;

#define D_FEAT 128

// ---------------------------------------------------------------------------
// Kernel 1: h = x  (eps = 0, so (1+eps)*x == x).  float4 vectorized copy.
// ---------------------------------------------------------------------------
__global__ void gin_init_h(const float4* __restrict__ x4,
                           float4* __restrict__ h4, int n4) {
    int i = blockIdx.x * blockDim.x + threadIdx.x;
    if (i < n4) h4[i] = x4[i];
}

// ---------------------------------------------------------------------------
// Kernel 2: scatter-add   h[dst] += x[src]  per edge.
// One wave (32 lanes) per edge; lane handles elements {lane, lane+32, +64, +96}
// so each atomic burst is a fully coalesced 128B cacheline touch.
// Inline asm guarantees the native fp32 atomic (no CAS loop) at device scope.
// ---------------------------------------------------------------------------
__device__ __forceinline__ void atomic_add_f32_dev(float* p, float v) {
    asm volatile("global_atomic_add_f32 %0, %1, off scope:SCOPE_DEV"
                 : : "v"(p), "v"(v) : "memory");
}

__global__ void gin_scatter_add(const float* __restrict__ x,
                                const int* __restrict__ edge_index,
                                float* __restrict__ h, int n_edges) {
    int t    = blockIdx.x * blockDim.x + threadIdx.x;
    int e    = t >> 5;
    int lane = t & 31;
    if (e >= n_edges) return;
    int src = edge_index[e];             // row 0 of [2, E]
    int dst = edge_index[n_edges + e];   // row 1 of [2, E]
    const float* xs = x + (size_t)src * D_FEAT;
    float*       hd = h + (size_t)dst * D_FEAT;
#pragma unroll
    for (int j = 0; j < 4; ++j) {
        atomic_add_f32_dev(hd + lane + 32 * j, xs[lane + 32 * j]);
    }
}

// ---------------------------------------------------------------------------
// Kernel 3: out = h @ W^T + b  via V_WMMA_F32_16X16X4_F32 (exact fp32 WMMA).
// One wave computes one 16x16 output tile (16 nodes x 16 out-features),
// looping K = 128 in steps of 4 (32 WMMA ops).
//
// fp32 A (16x4) layout: lane l -> row M = l&15; VGPR0/1 hold K = 2*(l>>4)+{0,1}
// fp32 B (4x16) layout: lane l -> col N = l&15; VGPR0/1 hold K = 2*(l>>4)+{0,1}
// Since B[k][n] = W[n][k], both A and B loads are contiguous float2 loads.
// C/D 16x16 f32: VGPR r -> (M = r + 8*(l>>4), N = l&15).
// ---------------------------------------------------------------------------
__global__ void gin_gemm_wmma(const float* __restrict__ h,
                              const float* __restrict__ W,
                              const float* __restrict__ bias,
                              float* __restrict__ out) {
    const int lane  = threadIdx.x & 31;
    const int wave  = threadIdx.x >> 5;                 // 8 waves / block
    const int tile  = blockIdx.x * 8 + wave;
    const int tileM = tile >> 3;                        // node-tile  (0..3124)
    const int tileN = tile & 7;                         // feat-tile  (0..7)

    const int half = lane >> 4;                         // 0 or 1
    const int lane16 = lane & 15;
    const int m = tileM * 16 + lane16;                  // node row for A
    const int n = tileN * 16 + lane16;                  // out col  for B

    const float* __restrict__ hrow = h + (size_t)m * D_FEAT;
    const float* __restrict__ wrow = W + (size_t)n * D_FEAT;

    v8f c = {};
#pragma unroll
    for (int kk = 0; kk < 32; ++kk) {
        const int k = kk * 4 + half * 2;
        v2f a = *(const v2f*)(hrow + k);
        v2f b = *(const v2f*)(wrow + k);
        // (neg_a, A, neg_b, B, c_mod, C, reuse_a, reuse_b)
        c = __builtin_amdgcn_wmma_f32_16x16x4_f32(
                false, a, false, b, (short)0, c, false, false);
    }

    const float bv   = bias[n];
    const int   col  = tileN * 16 + lane16;
    const int   row0 = tileM * 16 + half * 8;
#pragma unroll
    for (int r = 0; r < 8; ++r) {
        out[(size_t)(row0 + r) * D_FEAT + col] = c[r] + bv;
    }
}

// ---------------------------------------------------------------------------
extern "C" void kernel_launch(void* const* d_in, const int* in_sizes, int n_in,
                              void* d_out, int out_size, void* d_ws, size_t ws_size,
                              hipStream_t stream) {
    const float* x    = (const float*)d_in[0];
    const int*   ei   = (const int*)d_in[1];   // [2, E] int32
    const float* W    = (const float*)d_in[2]; // [128, 128] (out, in)
    const float* bias = (const float*)d_in[3]; // [128]
    float*       out  = (float*)d_out;
    float*       h    = (float*)d_ws;          // [n_nodes, 128] scratch

    const int n_nodes = in_sizes[0] / D_FEAT;  // 50000
    const int n_edges = in_sizes[1] / 2;       // 600000

    // 1) h = x
    {
        int n4 = n_nodes * (D_FEAT / 4);
        int blocks = (n4 + 255) / 256;
        gin_init_h<<<blocks, 256, 0, stream>>>((const float4*)x, (float4*)h, n4);
    }

    // 2) h[dst] += x[src] for each edge (one wave per edge)
    {
        long long threads = (long long)n_edges * 32;
        int blocks = (int)((threads + 255) / 256);
        gin_scatter_add<<<blocks, 256, 0, stream>>>(x, ei, h, n_edges);
    }

    // 3) out = h @ W^T + b   (WMMA fp32; 50000 % 16 == 0, 128/16 == 8 tiles)
    {
        int blocks = n_nodes / 16;  // 3125 blocks * 8 waves = all 16x16 tiles
        gin_gemm_wmma<<<blocks, 256, 0, stream>>>(h, W, bias, out);
    }
}